// GCNLayer_26534307955358
// MI455X (gfx1250) — compile-verified
//
#include <hip/hip_runtime.h>

typedef __attribute__((ext_vector_type(2))) float v2f;
typedef __attribute__((ext_vector_type(8))) float v8f;

#define BN_EPS_F 1e-5f
#define ATTN_SPLIT 8
#define ATTN_MAX_CHUNK 2048   // 2048*5*4 = 40 KB LDS

// ---------------------------------------------------------------------------
// BatchNorm: per-column stats (one block per column), then elementwise apply
// ---------------------------------------------------------------------------
__global__ __launch_bounds__(256) void bn_stats_kernel(
    const float* __restrict__ H, float* __restrict__ mu, float* __restrict__ rs,
    int n, int Cin) {
  __shared__ float s1[256];
  __shared__ float s2[256];
  const int c = blockIdx.x;
  float sum = 0.f, sq = 0.f;
  for (int r = threadIdx.x; r < n; r += 256) {
    float v = H[(size_t)r * Cin + c];
    sum += v;
    sq += v * v;
  }
  s1[threadIdx.x] = sum;
  s2[threadIdx.x] = sq;
  __syncthreads();
  for (int s = 128; s > 0; s >>= 1) {
    if (threadIdx.x < s) {
      s1[threadIdx.x] += s1[threadIdx.x + s];
      s2[threadIdx.x] += s2[threadIdx.x + s];
    }
    __syncthreads();
  }
  if (threadIdx.x == 0) {
    float m = s1[0] / (float)n;
    float var = s2[0] / (float)n - m * m;  // biased variance (training mode)
    mu[c] = m;
    rs[c] = rsqrtf(var + BN_EPS_F);
  }
}

__global__ __launch_bounds__(256) void bn_apply_kernel(
    const float* __restrict__ H, const float* __restrict__ mu,
    const float* __restrict__ rs, const float* __restrict__ gamma,
    const float* __restrict__ beta, float* __restrict__ Hn,
    int total, int Cin) {
  int idx = blockIdx.x * 256 + threadIdx.x;
  if (idx >= total) return;
  int c = idx % Cin;
  Hn[idx] = (H[idx] - mu[c]) * rs[c] * gamma[c] + beta[c];
}

// ---------------------------------------------------------------------------
// fp32 WMMA GEMM:  C[M,N] = op(A) * op(B) (+bias) (+relu)
//   BT=true : B is a weight matrix W[N,K] row-major (linear layer, B = W^T)
//   BT=false: B is [K,N] row-major
//   FUSED   : A := A + alpha * A2  (applied at use time, X never materialized)
//   EDGE    : N may not be a multiple of 16*NT -> clamp + select-0 (H_R only).
//   NT      : N-tiles per wave; one A fragment feeds NT back-to-back WMMAs.
// Block = 128 threads = 4 waves stacked in M. K must be a multiple of 4.
// Software-pipelined at the source level with #pragma unroll 1: each
// iteration issues the NEXT k-step's raw loads (one s_clause group), then
// runs the NT WMMAs on the CURRENT registers; the counted s_wait_loadcnt
// drains only the older load group while the next is in flight. Forcing
// unroll 1 keeps peak liveness at exactly 2 fragment sets (the compiler's
// own unroll-by-2 created a third set and spilled it into a serialized
// load->wait->wmma chain).
// Fragment layouts per CDNA5 ISA 7.12.2 (f32 16x16x4):
//   A 16x4: lanes 0-15 {K=k,k+1} M=lane; lanes 16-31 {K=k+2,k+3}
//   B 4x16: lanes 0-15 {K=k,k+1} N=lane; lanes 16-31 {K=k+2,k+3}
//   C 16x16: VGPR g -> M=g (lanes 0-15) / M=g+8 (lanes 16-31), N = lane&15
// ---------------------------------------------------------------------------
template <bool BT, bool FUSED, bool BIAS, bool RELU, bool EDGE, int NT>
__global__ __launch_bounds__(128) void wmma_gemm_kernel(
    const float* __restrict__ A, const float* __restrict__ A2,
    const float* __restrict__ alpha_p, const float* __restrict__ B,
    const float* __restrict__ bias, float* __restrict__ C,
    int M, int N, int K) {
  const int lane = threadIdx.x & 31;
  const int wave = threadIdx.x >> 5;
  const int half = lane >> 4;   // 0: K pair (k,k+1), 1: K pair (k+2,k+3)
  const int l16  = lane & 15;
  const int tm  = blockIdx.x * 64 + wave * 16;
  const int tn0 = blockIdx.y * (16 * NT);
  const int am  = tm + l16;
  const float alpha = FUSED ? alpha_p[0] : 0.f;

  // Per-lane loop-invariant element offsets (32-bit).
  const int laneA = am * K + half * 2;

  bool ok[NT];
  int laneB[NT];   // per-lane element offset for tile t (clamped if EDGE)
#pragma unroll
  for (int t = 0; t < NT; ++t) {
    int bn = tn0 + t * 16 + l16;
    ok[t] = EDGE ? (bn < N) : true;
    int bnc = (EDGE && !ok[t]) ? (N - 1) : bn;
    laneB[t] = BT ? (bnc * K + half * 2)   // row of W + K-pair
                  : (bnc + half * 2 * N);  // col of B + K-pair*row stride
  }

  v8f acc[NT];
#pragma unroll
  for (int t = 0; t < NT; ++t) acc[t] = (v8f){};

  // ---- raw-load helpers: index = per-lane invariant + uniform k part ----
  auto loadA = [&](int k, v2f& ra, v2f& ra2) {
    ra.x = A[laneA + k];
    ra.y = A[laneA + k + 1];
    if (FUSED) {
      ra2.x = A2[laneA + k];
      ra2.y = A2[laneA + k + 1];
    }
  };
  auto loadB = [&](int t, int k, v2f& rb) {
    if (BT) {
      rb.x = B[laneB[t] + k];
      rb.y = B[laneB[t] + k + 1];
    } else {
      rb.x = B[laneB[t] + k * N];
      rb.y = B[laneB[t] + k * N + N];
    }
  };
  auto makeA = [&](const v2f& ra, const v2f& ra2) -> v2f {
    v2f a;
    if (FUSED) {
      a.x = fmaf(alpha, ra2.x, ra.x);
      a.y = fmaf(alpha, ra2.y, ra.y);
    } else {
      a = ra;
    }
    return a;
  };
  auto makeB = [&](int t, const v2f& rb) -> v2f {
    if (!EDGE) return rb;             // no masking code at all
    v2f b;
    b.x = ok[t] ? rb.x : 0.f;         // branchless select (address was safe)
    b.y = ok[t] ? rb.y : 0.f;
    return b;
  };

  // ---- software pipeline: prologue loads k=0 ----
  v2f ra_c, ra2_c, rb_c[NT];
  v2f ra_n, ra2_n, rb_n[NT];
  loadA(0, ra_c, ra2_c);
#pragma unroll
  for (int t = 0; t < NT; ++t) loadB(t, 0, rb_c[t]);

#pragma unroll 1
  for (int k = 0; k + 4 < K; k += 4) {
    // issue next k-step's loads (batched; no dependent VALU yet)
    loadA(k + 4, ra_n, ra2_n);
#pragma unroll
    for (int t = 0; t < NT; ++t) loadB(t, k + 4, rb_n[t]);

    // compute with current registers (covers latency of the in-flight loads)
    v2f a = makeA(ra_c, ra2_c);
#pragma unroll
    for (int t = 0; t < NT; ++t) {
      v2f b = makeB(t, rb_c[t]);
      acc[t] = __builtin_amdgcn_wmma_f32_16x16x4_f32(
          false, a, false, b, (short)0, acc[t], false, false);
    }

    ra_c = ra_n;
    ra2_c = ra2_n;
#pragma unroll
    for (int t = 0; t < NT; ++t) rb_c[t] = rb_n[t];
  }
  {  // epilogue of the pipeline (last k-step)
    v2f a = makeA(ra_c, ra2_c);
#pragma unroll
    for (int t = 0; t < NT; ++t) {
      v2f b = makeB(t, rb_c[t]);
      acc[t] = __builtin_amdgcn_wmma_f32_16x16x4_f32(
          false, a, false, b, (short)0, acc[t], false, false);
    }
  }

  const int rbase = tm + half * 8;
#pragma unroll
  for (int t = 0; t < NT; ++t) {
    int col = tn0 + t * 16 + l16;
    if (!EDGE || col < N) {
      float bv = BIAS ? bias[col] : 0.f;
#pragma unroll
      for (int g = 0; g < 8; ++g) {
        float v = acc[t][g] + bv;
        if (RELU) v = fmaxf(v, 0.f);
        C[(size_t)(rbase + g) * N + col] = v;
      }
    }
  }
}

// ---------------------------------------------------------------------------
// aenet head: V[n,CN] -> relu(lin 10) -> relu(lin 5) = V2[n,5]
// ae_w1 (10 x CN) staged in LDS; V1 lives in registers only.
// ---------------------------------------------------------------------------
__global__ __launch_bounds__(256) void aenet12_kernel(
    const float* __restrict__ V, const float* __restrict__ w1,
    const float* __restrict__ b1, const float* __restrict__ w2,
    const float* __restrict__ b2, float* __restrict__ V2, int n, int CN) {
  __shared__ float sw1[10 * 256];   // CN <= 256
  for (int t = threadIdx.x; t < 10 * CN; t += 256) sw1[t] = w1[t];
  __syncthreads();

  int i = blockIdx.x * 256 + threadIdx.x;
  if (i >= n) return;
  float acc[10];
#pragma unroll
  for (int c = 0; c < 10; ++c) acc[c] = b1[c];
  const float* vr = V + (size_t)i * CN;
  for (int k = 0; k < CN; ++k) {
    float v = vr[k];
#pragma unroll
    for (int c = 0; c < 10; ++c) acc[c] = fmaf(v, sw1[c * CN + k], acc[c]);
  }
#pragma unroll
  for (int c = 0; c < 10; ++c) acc[c] = fmaxf(acc[c], 0.f);
#pragma unroll
  for (int c = 0; c < 5; ++c) {
    float t = b2[c];
#pragma unroll
    for (int k = 0; k < 10; ++k) t = fmaf(w2[c * 10 + k], acc[k], t);
    V2[(size_t)i * 5 + c] = fmaxf(t, 0.f);
  }
}

// ---------------------------------------------------------------------------
// Flash-style attention, d=5: softmax(V2 V2^T) V2, split ATTN_SPLIT ways over
// the key axis. Each block stages its key chunk in LDS; each thread runs an
// online softmax over the chunk for one query row. Partials (m, l, acc[5])
// are merged by attn_combine via log-sum-exp.
// ---------------------------------------------------------------------------
__global__ __launch_bounds__(256) void attn_part_kernel(
    const float* __restrict__ V2, float* __restrict__ P, int n) {
  __shared__ float lv[ATTN_MAX_CHUNK * 5];
  const int chunk = n / ATTN_SPLIT;
  const int blocksPerPart = (n + 255) / 256;
  const int part = blockIdx.x / blocksPerPart;
  const int i = (blockIdx.x % blocksPerPart) * 256 + threadIdx.x;
  const int j0 = part * chunk;

  for (int t = threadIdx.x; t < chunk * 5; t += 256)
    lv[t] = V2[(size_t)j0 * 5 + t];
  __syncthreads();
  if (i >= n) return;

  const float q0 = V2[(size_t)i * 5 + 0];
  const float q1 = V2[(size_t)i * 5 + 1];
  const float q2 = V2[(size_t)i * 5 + 2];
  const float q3 = V2[(size_t)i * 5 + 3];
  const float q4 = V2[(size_t)i * 5 + 4];

  float m = -3.0e38f, l = 0.f;
  float a0 = 0.f, a1 = 0.f, a2 = 0.f, a3 = 0.f, a4 = 0.f;
  for (int j = 0; j < chunk; ++j) {
    const float* kv = &lv[j * 5];
    float s = q0 * kv[0] + q1 * kv[1] + q2 * kv[2] + q3 * kv[3] + q4 * kv[4];
    float mn = fmaxf(m, s);
    float corr = __expf(m - mn);
    float p = __expf(s - mn);
    l = l * corr + p;
    a0 = fmaf(p, kv[0], a0 * corr);
    a1 = fmaf(p, kv[1], a1 * corr);
    a2 = fmaf(p, kv[2], a2 * corr);
    a3 = fmaf(p, kv[3], a3 * corr);
    a4 = fmaf(p, kv[4], a4 * corr);
    m = mn;
  }
  float* out = P + ((size_t)part * n + i) * 8;  // 8-float stride for alignment
  out[0] = m; out[1] = l;
  out[2] = a0; out[3] = a1; out[4] = a2; out[5] = a3; out[6] = a4;
}

__global__ __launch_bounds__(256) void attn_combine_kernel(
    const float* __restrict__ P, float* __restrict__ V3, int n) {
  int i = blockIdx.x * 256 + threadIdx.x;
  if (i >= n) return;
  float M = -3.0e38f;
  for (int p = 0; p < ATTN_SPLIT; ++p)
    M = fmaxf(M, P[((size_t)p * n + i) * 8]);
  float L = 0.f, a[5] = {0.f, 0.f, 0.f, 0.f, 0.f};
  for (int p = 0; p < ATTN_SPLIT; ++p) {
    const float* b = P + ((size_t)p * n + i) * 8;
    float w = __expf(b[0] - M);
    L = fmaf(b[1], w, L);
#pragma unroll
    for (int d = 0; d < 5; ++d) a[d] = fmaf(b[2 + d], w, a[d]);
  }
  float inv = 1.f / L;
#pragma unroll
  for (int d = 0; d < 5; ++d) V3[(size_t)i * 5 + d] = a[d] * inv;
}

// ---------------------------------------------------------------------------
// aenet tail: V3[n,5] -> relu(lin 10) = V4[n,10], then per-element (i,c):
// S[i,c] = relu(b4[c] + sum_k V4[i,k] * w4[c,k])   (K=10 -> plain VALU)
// ---------------------------------------------------------------------------
__global__ __launch_bounds__(256) void ae3_kernel(
    const float* __restrict__ V3, const float* __restrict__ w3,
    const float* __restrict__ b3, float* __restrict__ V4, int n) {
  int i = blockIdx.x * 256 + threadIdx.x;
  if (i >= n) return;
  float v[5];
#pragma unroll
  for (int d = 0; d < 5; ++d) v[d] = V3[(size_t)i * 5 + d];
#pragma unroll
  for (int c = 0; c < 10; ++c) {
    float t = b3[c];
#pragma unroll
    for (int k = 0; k < 5; ++k) t = fmaf(w3[c * 5 + k], v[k], t);
    V4[(size_t)i * 10 + c] = fmaxf(t, 0.f);
  }
}

__global__ __launch_bounds__(256) void ae4_kernel(
    const float* __restrict__ V4, const float* __restrict__ w4,
    const float* __restrict__ b4, float* __restrict__ S, int n, int CN) {
  int idx = blockIdx.x * 256 + threadIdx.x;
  if (idx >= n * CN) return;
  int i = idx / CN;
  int c = idx % CN;
  float t = b4[c];
#pragma unroll
  for (int k = 0; k < 10; ++k)
    t = fmaf(V4[(size_t)i * 10 + k], w4[c * 10 + k], t);
  S[idx] = fmaxf(t, 0.f);
}

// ---------------------------------------------------------------------------
// Host-side orchestration
// ---------------------------------------------------------------------------
static void run_aenet(const float* V, float* V2b, float* Pb, float* V3b,
                      float* V4b, float* S,
                      const float* aw1, const float* ab1, const float* aw2,
                      const float* ab2, const float* aw3, const float* ab3,
                      const float* aw4, const float* ab4,
                      int n, int CN, hipStream_t stream) {
  int rb = (n + 255) / 256;
  aenet12_kernel<<<rb, 256, 0, stream>>>(V, aw1, ab1, aw2, ab2, V2b, n, CN);
  attn_part_kernel<<<rb * ATTN_SPLIT, 256, 0, stream>>>(V2b, Pb, n);
  attn_combine_kernel<<<rb, 256, 0, stream>>>(Pb, V3b, n);
  ae3_kernel<<<rb, 256, 0, stream>>>(V3b, aw3, ab3, V4b, n);
  ae4_kernel<<<(n * CN + 255) / 256, 256, 0, stream>>>(V4b, aw4, ab4, S, n, CN);
}

#define GEMM_NT 8

extern "C" void kernel_launch(void* const* d_in, const int* in_sizes, int n_in,
                              void* d_out, int out_size, void* d_ws, size_t ws_size,
                              hipStream_t stream) {
  const float* H      = (const float*)d_in[0];
  const float* alpha  = (const float*)d_in[1];
  const float* gamma  = (const float*)d_in[2];
  const float* beta   = (const float*)d_in[3];
  const float* w1     = (const float*)d_in[4];
  const float* b1     = (const float*)d_in[5];
  const float* wdt    = (const float*)d_in[6];
  const float* bdt    = (const float*)d_in[7];
  const float* wdtd   = (const float*)d_in[8];
  const float* bdtd   = (const float*)d_in[9];
  const float* aw1    = (const float*)d_in[10];
  const float* ab1    = (const float*)d_in[11];
  const float* aw2    = (const float*)d_in[12];
  const float* ab2    = (const float*)d_in[13];
  const float* aw3    = (const float*)d_in[14];
  const float* ab3    = (const float*)d_in[15];
  const float* aw4    = (const float*)d_in[16];
  const float* ab4    = (const float*)d_in[17];

  const int Cin = in_sizes[2];          // 200
  const int CN  = in_sizes[5];          // 128
  const int n   = in_sizes[0] / Cin;    // 8192

  // Workspace carve-up (floats)
  float* ws = (float*)d_ws;
  float* mu  = ws;                 ws += 256;            // Cin <= 256
  float* rs  = ws;                 ws += 256;
  float* Hn  = ws;                 ws += (size_t)n * Cin;
  float* Vb  = ws;                 ws += (size_t)n * CN;
  float* DTH = ws;                 ws += (size_t)n * CN;
  float* S   = ws;                 ws += (size_t)n * CN;
  float* V2b = ws;                 ws += (size_t)n * 5;
  float* V3b = ws;                 ws += (size_t)n * 5;
  float* V4b = ws;                 ws += (size_t)n * 10;
  float* Pb  = ws;                 ws += (size_t)ATTN_SPLIT * n * 8;
  (void)ws_size; (void)n_in;

  float* Vout = (float*)d_out;                    // [n, CN]
  float* HR   = (float*)d_out + (size_t)n * CN;   // [n, Cin]

  // 1) BatchNorm
  bn_stats_kernel<<<Cin, 256, 0, stream>>>(H, mu, rs, n, Cin);
  bn_apply_kernel<<<(n * Cin + 255) / 256, 256, 0, stream>>>(
      H, mu, rs, gamma, beta, Hn, n * Cin, Cin);

  dim3 gB(128, 1, 1);
  const int strip = 16 * GEMM_NT;
  dim3 gGemmCN(n / 64, (CN + strip - 1) / strip);     // N=128 -> exact strip
  dim3 gGemmCin(n / 64, (Cin + strip - 1) / strip);   // N=200 -> edge tiles

  // 2) V = relu(Hn @ w1^T + b1);  DTH = Hn @ wdt^T + bdt   (no edge: N==128)
  wmma_gemm_kernel<true, false, true, true, false, GEMM_NT>
      <<<gGemmCN, gB, 0, stream>>>(Hn, nullptr, nullptr, w1, b1, Vb, n, CN, Cin);
  wmma_gemm_kernel<true, false, true, false, false, GEMM_NT>
      <<<gGemmCN, gB, 0, stream>>>(Hn, nullptr, nullptr, wdt, bdt, DTH, n, CN, Cin);

  // 3) S = relu(aenet(V))
  run_aenet(Vb, V2b, Pb, V3b, V4b, S, aw1, ab1, aw2, ab2, aw3, ab3, aw4, ab4,
            n, CN, stream);

  // 4) Three update iterations; last aenet is dead code (outputs use V only).
  for (int it = 0; it < 3; ++it) {
    float* Vdst = (it == 2) ? Vout : Vb;
    wmma_gemm_kernel<true, true, true, true, false, GEMM_NT>
        <<<gGemmCN, gB, 0, stream>>>(DTH, S, alpha, wdtd, bdtd, Vdst, n, CN, CN);
    if (it < 2)
      run_aenet(Vdst, V2b, Pb, V3b, V4b, S, aw1, ab1, aw2, ab2, aw3, ab3,
                aw4, ab4, n, CN, stream);
  }

  // 5) H_R = V @ wdt  (B non-transposed [CN, Cin]; N=200 -> EDGE masking on)
  wmma_gemm_kernel<false, false, false, false, true, GEMM_NT>
      <<<gGemmCin, gB, 0, stream>>>(Vout, nullptr, nullptr, wdt, nullptr, HR,
                                    n, Cin, CN);
}